// GCN_68848325755002
// MI455X (gfx1250) — compile-verified
//
#include <hip/hip_runtime.h>
#include <hip/hip_bf16.h>

// ---------------- problem constants ----------------
#define NN      8192
#define F_IN    128
#define OC      64        // OC1 == OC2 == 64
#define NCLASS  16
#define CAP     128       // neighbor capacity per row (avg deg ~33; Poisson tail @128 ~ 0)

typedef float  f4  __attribute__((ext_vector_type(4)));
typedef float  v2f __attribute__((ext_vector_type(2)));
typedef float  v8f __attribute__((ext_vector_type(8)));

// ---------------------------------------------------------------------------
// Kernel 1: scan dense A (256 MB, streamed ONCE, non-temporal) and build a
// fixed-capacity CSR: nbr[row][k], cnt[row], firstm[row] = min nonzero column.
// One workgroup per row; LDS atomics for compaction (order-insensitive sum).
// ---------------------------------------------------------------------------
__global__ void k_build_csr(const float* __restrict__ A,
                            int* __restrict__ nbr,
                            int* __restrict__ cnt,
                            int* __restrict__ firstm) {
    const int row = blockIdx.x;
    __shared__ int lcnt, lmin;
    if (threadIdx.x == 0) { lcnt = 0; lmin = 0x7FFFFFFF; }
    __syncthreads();
    const float* Ar = A + (size_t)row * NN;
    for (int c0 = threadIdx.x * 4; c0 < NN; c0 += 256 * 4) {
        f4 v = __builtin_nontemporal_load((const f4*)(Ar + c0));   // TH_NT stream
        #pragma unroll
        for (int u = 0; u < 4; ++u) {
            if (v[u] != 0.0f) {
                int pos = atomicAdd(&lcnt, 1);
                if (pos < CAP) nbr[(size_t)row * CAP + pos] = c0 + u;
                atomicMin(&lmin, c0 + u);
            }
        }
    }
    __syncthreads();
    if (threadIdx.x == 0) {
        cnt[row]    = lcnt < CAP ? lcnt : CAP;
        firstm[row] = lmin;   // self-loop guarantees a nonzero
    }
}

// ---------------------------------------------------------------------------
// Kernel 2: deg from diagonal of D (== jnp.max(D, axis=1) since deg >= 1).
// invs[j] = rsqrt(deg[j]);  rowscale[i] = rsqrt(deg[firstm[i]]).
// ---------------------------------------------------------------------------
__global__ void k_norm(const float* __restrict__ D,
                       const int* __restrict__ firstm,
                       float* __restrict__ invs,
                       float* __restrict__ rowscale) {
    int i = blockIdx.x * blockDim.x + threadIdx.x;
    if (i >= NN) return;
    float d  = D[(size_t)i * NN + i];
    invs[i]  = 1.0f / sqrtf(d);
    int   fi = firstm[i];
    float df = D[(size_t)fi * NN + fi];
    rowscale[i] = 1.0f / sqrtf(df);
}

// ---------------------------------------------------------------------------
// Kernel 3: Hs[j,:] = invs[j] * H[j,:]   (column-normalization folded in)
// ---------------------------------------------------------------------------
__global__ void k_scale(const float* __restrict__ Hin,
                        const float* __restrict__ invs,
                        float* __restrict__ Hout, int F) {
    int idx = blockIdx.x * blockDim.x + threadIdx.x;
    if (idx >= NN * F) return;
    Hout[idx] = Hin[idx] * invs[idx / F];
}

// ---------------------------------------------------------------------------
// Kernel 4: sparse aggregation  agg[i,:] = rowscale[i] * sum_k Hs[nbr[i,k],:]
// blockDim.x == F; neighbor list staged in LDS; feature loads coalesced;
// prefetch next neighbor row (global_prefetch_b8).
// ---------------------------------------------------------------------------
__global__ void k_spmm(const int* __restrict__ nbr,
                       const int* __restrict__ cnt,
                       const float* __restrict__ Hs,
                       const float* __restrict__ rowscale,
                       float* __restrict__ agg, int F) {
    const int row = blockIdx.x;
    const int f   = threadIdx.x;
    __shared__ int snbr[CAP];
    const int n = cnt[row];
    for (int k = f; k < n; k += F) snbr[k] = nbr[(size_t)row * CAP + k];
    __syncthreads();
    float acc = 0.0f;
    for (int k = 0; k < n; ++k) {
        int j = snbr[k];
        if (k + 1 < n)
            __builtin_prefetch(&Hs[(size_t)snbr[k + 1] * F + f], 0, 0);
        acc += Hs[(size_t)j * F + f];
    }
    agg[(size_t)row * F + f] = rowscale[row] * acc;
}

// ---------------------------------------------------------------------------
// Kernel 5: dense transform  Out = leaky_relu(Ag @ Wl + bl), Out is [8192, 64]
// via V_WMMA_F32_16X16X4_F32.  128 threads = 4 waves; workgroup owns a 16-row
// M-tile; wave w owns output columns [16w, 16w+16).
//   A frag (16x4 f32): lane L (m=L&15, hi=L>>4) holds A[m][kk+2hi], A[m][kk+2hi+1]
//   B frag (4x16 f32): lane L (n=L&15)          holds B[kk+2hi][n], B[kk+2hi+1][n]
//   C/D (16x16 f32):   VGPR r, lane L -> D[r+8hi][n]
// ---------------------------------------------------------------------------
template <int K>
__global__ void k_gemm_wmma(const float* __restrict__ Ag,
                            const float* __restrict__ Wl,
                            const float* __restrict__ bl,
                            float* __restrict__ Out) {
    constexpr int LDA = K + 1;                 // pad: stride K+1 -> conflict-free LDS
    __shared__ float sA[16 * LDA];
    __shared__ float sW[K * OC];
    const int row0 = blockIdx.x * 16;
    const int tid  = threadIdx.x;              // 0..127

    for (int idx = tid; idx < 16 * K; idx += 128) {
        int r = idx / K, c = idx % K;
        sA[r * LDA + c] = Ag[(size_t)(row0 + r) * K + c];
    }
    for (int idx = tid; idx < K * OC; idx += 128) sW[idx] = Wl[idx];
    __syncthreads();

    const int lane = tid & 31;
    const int m    = lane & 15;                // M for A-frag, N for B-frag/D
    const int hi   = lane >> 4;
    const int nb   = (tid >> 5) * 16;          // wave's N-tile base

    v8f c = {0.f, 0.f, 0.f, 0.f, 0.f, 0.f, 0.f, 0.f};
    #pragma unroll
    for (int kk = 0; kk < K; kk += 4) {
        v2f a, b;
        a.x = sA[m * LDA + kk + 2 * hi];
        a.y = sA[m * LDA + kk + 2 * hi + 1];
        b.x = sW[(kk + 2 * hi)     * OC + nb + m];
        b.y = sW[(kk + 2 * hi + 1) * OC + nb + m];
        c = __builtin_amdgcn_wmma_f32_16x16x4_f32(
                /*neg_a=*/false, a, /*neg_b=*/false, b,
                /*c_mod=*/(short)0, c, /*reuse_a=*/false, /*reuse_b=*/false);
    }

    const float bias = bl[nb + m];
    #pragma unroll
    for (int r = 0; r < 8; ++r) {
        const int mm = r + 8 * hi;
        float v = c[r] + bias;
        v = v > 0.0f ? v : 0.01f * v;          // leaky_relu(0.01)
        Out[(size_t)(row0 + mm) * OC + nb + m] = v;
    }
}

// ---------------------------------------------------------------------------
// Kernel 6: logits = H2 @ W3 + b3 ; log_softmax over the 16 classes.
// ---------------------------------------------------------------------------
__global__ void k_head(const float* __restrict__ H,
                       const float* __restrict__ W3,
                       const float* __restrict__ b3,
                       float* __restrict__ out) {
    __shared__ float sW[OC * NCLASS];
    __shared__ float sb[NCLASS];
    const int tid = threadIdx.x;
    for (int i = tid; i < OC * NCLASS; i += 256) sW[i] = W3[i];
    if (tid < NCLASS) sb[tid] = b3[tid];
    __syncthreads();
    const int row = blockIdx.x * 256 + tid;
    if (row >= NN) return;
    const float* Hr = H + (size_t)row * OC;
    float h[OC];
    #pragma unroll
    for (int k = 0; k < OC; ++k) h[k] = Hr[k];
    float lg[NCLASS];
    #pragma unroll
    for (int c = 0; c < NCLASS; ++c) {
        float s = sb[c];
        #pragma unroll
        for (int k = 0; k < OC; ++k) s += h[k] * sW[k * NCLASS + c];
        lg[c] = s;
    }
    float mx = lg[0];
    #pragma unroll
    for (int c = 1; c < NCLASS; ++c) mx = fmaxf(mx, lg[c]);
    float se = 0.0f;
    #pragma unroll
    for (int c = 0; c < NCLASS; ++c) se += expf(lg[c] - mx);
    const float lse = mx + logf(se);
    #pragma unroll
    for (int c = 0; c < NCLASS; ++c) out[(size_t)row * NCLASS + c] = lg[c] - lse;
}

// ---------------------------------------------------------------------------
// inputs (setup_inputs order): 0:D 1:X 2:A 3:W1 4:b1 5:W2 6:b2 7:W3 8:b3
// ---------------------------------------------------------------------------
extern "C" void kernel_launch(void* const* d_in, const int* in_sizes, int n_in,
                              void* d_out, int out_size, void* d_ws, size_t ws_size,
                              hipStream_t stream) {
    (void)in_sizes; (void)n_in; (void)out_size; (void)ws_size;
    const float* D  = (const float*)d_in[0];
    const float* X  = (const float*)d_in[1];
    const float* A  = (const float*)d_in[2];
    const float* W1 = (const float*)d_in[3];
    const float* b1 = (const float*)d_in[4];
    const float* W2 = (const float*)d_in[5];
    const float* b2 = (const float*)d_in[6];
    const float* W3 = (const float*)d_in[7];
    const float* b3 = (const float*)d_in[8];
    float* out = (float*)d_out;

    // workspace layout (bytes), total ~14.8 MB
    char* ws = (char*)d_ws;
    int*   nbr      = (int*)  (ws);                                   // 8192*128*4
    int*   cnt      = (int*)  (ws + (size_t)NN * CAP * 4);            // 32 KB
    int*   firstm   = (int*)  (ws + (size_t)NN * CAP * 4 + 32768);
    float* invs     = (float*)(ws + (size_t)NN * CAP * 4 + 2 * 32768);
    float* rowscale = (float*)(ws + (size_t)NN * CAP * 4 + 3 * 32768);
    float* Hs       = (float*)(ws + (size_t)NN * CAP * 4 + 4 * 32768);          // 4 MB
    float* agg      = (float*)((char*)Hs + (size_t)NN * F_IN * 4);              // 4 MB
    float* H        = (float*)((char*)agg + (size_t)NN * F_IN * 4);             // 2 MB

    // 1) sparse structure from A (single 256 MB streaming read — the roofline)
    k_build_csr<<<NN, 256, 0, stream>>>(A, nbr, cnt, firstm);
    // 2) normalization terms from diag(D)
    k_norm<<<NN / 256, 256, 0, stream>>>(D, firstm, invs, rowscale);

    // layer 1: Hs = invs*X ; agg = rowscale * gather-sum ; H1 = lrelu(agg@W1+b1)
    k_scale<<<(NN * F_IN) / 256, 256, 0, stream>>>(X, invs, Hs, F_IN);
    k_spmm<<<NN, F_IN, 0, stream>>>(nbr, cnt, Hs, rowscale, agg, F_IN);
    k_gemm_wmma<F_IN><<<NN / 16, 128, 0, stream>>>(agg, W1, b1, H);

    // layer 2 (dropout in eval mode == identity)
    k_scale<<<(NN * OC) / 256, 256, 0, stream>>>(H, invs, Hs, OC);
    k_spmm<<<NN, OC, 0, stream>>>(nbr, cnt, Hs, rowscale, agg, OC);
    k_gemm_wmma<OC><<<NN / 16, 128, 0, stream>>>(agg, W2, b2, H);

    // head: logits + log_softmax -> d_out [8192,16]
    k_head<<<NN / 256, 256, 0, stream>>>(H, W3, b3, out);
}